// MNISTV3Classifier_43817256354070
// MI455X (gfx1250) — compile-verified
//
#include <hip/hip_runtime.h>

typedef float v2f __attribute__((ext_vector_type(2)));
typedef float v8f __attribute__((ext_vector_type(8)));

#define WAVES_PER_BLOCK 4
#define DT_C 0.1f

// One 16x16 f32 D-tile of  W(rows m0..m0+15, K cols) @ Xt(K x 16), K accumulation
// via V_WMMA_F32_16X16X4_F32 (M=16,N=16,K=4).
// A layout: lane l holds row M=m0+(l%16); comp c -> K = kt + 2*(l/16) + c  (float2 global load)
// B layout: lane l holds col N=(l%16);   comp c -> K = kt + 2*(l/16) + c  (two LDS b32 loads)
__device__ __forceinline__ v8f wmma_accum_tile(const float* __restrict__ Wrow,
                                               const float* __restrict__ Xt,
                                               int K, int h, int li) {
  v8f acc = {0.f, 0.f, 0.f, 0.f, 0.f, 0.f, 0.f, 0.f};
  for (int kt = 0; kt < K; kt += 4) {
    v2f a = *(const v2f*)(Wrow + kt + 2 * h);      // 8B aligned: even element offsets
    v2f b;
    b.x = Xt[(kt + 2 * h) * 16 + li];
    b.y = Xt[(kt + 2 * h + 1) * 16 + li];
    acc = __builtin_amdgcn_wmma_f32_16x16x4_f32(false, a, false, b, (short)0, acc,
                                                false, false);
  }
  return acc;
}

__global__ void __launch_bounds__(WAVES_PER_BLOCK * 32)
mnist_v3_kernel(const float* __restrict__ image, const float* __restrict__ W_ff,
                const float* __restrict__ W_fb, const float* __restrict__ W_vel,
                const float* __restrict__ Wc1, const float* __restrict__ bc1,
                const float* __restrict__ Wc2, const float* __restrict__ bc2,
                float* __restrict__ out) {
  // Per-wave feature block [128 rows x 16 imgs]: rows 0-31 y_ff | 32-63 y_fb | 64-95 r6 | 96-127 |eps|
  __shared__ float sF[WAVES_PER_BLOCK][128 * 16];   // 32 KB
  __shared__ float sTmp[WAVES_PER_BLOCK][64 * 16];  // 16 KB: S(32x16)+E(32x16), later H(64x16)

  const int tid  = threadIdx.x;
  const int wave = tid >> 5;
  const int lane = tid & 31;
  const int h    = lane >> 4;   // half-wave select
  const int li   = lane & 15;   // image column / N index

  float* Fblk = sF[wave];
  float* YFF  = Fblk;
  float* YFB  = Fblk + 32 * 16;
  float* R    = Fblk + 64 * 16;
  float* AB   = Fblk + 96 * 16;
  float* S    = sTmp[wave];            // sensory [32][16]
  float* E    = sTmp[wave] + 32 * 16;  // topdown, then eps [32][16]
  float* H    = sTmp[wave];            // hidden [64][16], aliases S/E after they die

  const int imgBase = blockIdx.x * (WAVES_PER_BLOCK * 16) + wave * 16;
  const float* img  = image + (size_t)(imgBase + li) * 784;

  // ---- gather + linear interp (each lane: 16 features of its image) ----
#pragma unroll
  for (int j0 = 0; j0 < 16; ++j0) {
    int   j   = h * 16 + j0;
    float pos = (j + 0.5f) * 24.5f - 0.5f;   // in [11.75, 771.25], no clamping needed
    int   lo  = (int)pos;
    float w   = pos - (float)lo;
    float a0  = img[lo];
    float a1  = img[lo + 1];
    S[j * 16 + li] = (a0 * (1.0f - w) + a1 * w) * 80.0f;
    int p0 = lo, p1 = lo + 1;                 // transposed-flatten gather
    float t0 = img[(p0 % 28) * 28 + p0 / 28];
    float t1 = img[(p1 % 28) * 28 + p1 / 28];
    E[j * 16 + li] = (t0 * (1.0f - w) + t1 * w) * 40.0f;
    R[j * 16 + li] = 0.0f;
  }
  __syncthreads();

  // ---- y_fb = relu(W_fb @ topdown^T) ----
#pragma unroll
  for (int m0 = 0; m0 < 32; m0 += 16) {
    v8f d = wmma_accum_tile(W_fb + (m0 + li) * 32, E, 32, h, li);
#pragma unroll
    for (int r = 0; r < 8; ++r) {
      int m = m0 + r + 8 * h;
      YFB[m * 16 + li] = fmaxf(d[r], 0.0f);
    }
  }
  __syncthreads();

  const float VX[3] = {0.2f, 0.0f, 0.15f};
  const float VY[3] = {0.0f, 0.2f, 0.15f};

  for (int vi = 0; vi < 3; ++vi) {
    float vx = VX[vi], vy = VY[vi];
    for (int ps = 0; ps < 3; ++ps) {
      // E = S - W_fb @ R        (eps_pos - eps_neg == sensory - pred)
#pragma unroll
      for (int m0 = 0; m0 < 32; m0 += 16) {
        v8f d = wmma_accum_tile(W_fb + (m0 + li) * 32, R, 32, h, li);
#pragma unroll
        for (int r = 0; r < 8; ++r) {
          int m = m0 + r + 8 * h;
          E[m * 16 + li] = S[m * 16 + li] - d[r];
        }
      }
      __syncthreads();
      // YFF = relu(W_ff @ E)
#pragma unroll
      for (int m0 = 0; m0 < 32; m0 += 16) {
        v8f d = wmma_accum_tile(W_ff + (m0 + li) * 32, E, 32, h, li);
#pragma unroll
        for (int r = 0; r < 8; ++r) {
          int m = m0 + r + 8 * h;
          YFF[m * 16 + li] = fmaxf(d[r], 0.0f);
        }
      }
      __syncthreads();
      // r6 leaky integration
#pragma unroll
      for (int f0 = 0; f0 < 16; ++f0) {
        int f = h * 16 + f0;
        v2f wv = *(const v2f*)(W_vel + f * 2);
        float velin = wv.x * vx + wv.y * vy;
        float rv = R[f * 16 + li];
        R[f * 16 + li] =
            rv + DT_C * (-rv + YFF[f * 16 + li] + YFB[f * 16 + li] + velin);
      }
      __syncthreads();
    }
  }

  // ---- |eps| block of the feature ----
#pragma unroll
  for (int f0 = 0; f0 < 16; ++f0) {
    int f = h * 16 + f0;
    AB[f * 16 + li] = fabsf(E[f * 16 + li]);
  }
  __syncthreads();

  // ---- H = gelu_exact(Wc1 @ F + bc1)  [64 x 16] ----
#pragma unroll
  for (int m0 = 0; m0 < 64; m0 += 16) {
    v8f d = wmma_accum_tile(Wc1 + (m0 + li) * 128, Fblk, 128, h, li);
#pragma unroll
    for (int r = 0; r < 8; ++r) {
      int m = m0 + r + 8 * h;
      float x = d[r] + bc1[m];
      H[m * 16 + li] = 0.5f * x * (1.0f + erff(x * 0.70710678118654752f));
    }
  }
  __syncthreads();

  // ---- logits = Wc2 @ H + bc2  (10 rows; pad A rows by clamping, store only m<10) ----
  {
    int wrow = li < 10 ? li : 9;  // avoid OOB reads of Wc2; rows >=10 discarded
    v8f d = wmma_accum_tile(Wc2 + wrow * 64, H, 64, h, li);
    float* orow = out + (size_t)(imgBase + li) * 10;
#pragma unroll
    for (int r = 0; r < 8; ++r) {
      int m = r + 8 * h;
      if (m < 10) orow[m] = d[r] + bc2[m];
    }
  }
}

extern "C" void kernel_launch(void* const* d_in, const int* in_sizes, int n_in,
                              void* d_out, int out_size, void* d_ws, size_t ws_size,
                              hipStream_t stream) {
  const float* image = (const float*)d_in[0];
  const float* W_ff  = (const float*)d_in[1];
  const float* W_fb  = (const float*)d_in[2];
  const float* W_vel = (const float*)d_in[3];
  const float* Wc1   = (const float*)d_in[4];
  const float* bc1   = (const float*)d_in[5];
  const float* Wc2   = (const float*)d_in[6];
  const float* bc2   = (const float*)d_in[7];
  float* out = (float*)d_out;

  int B = in_sizes[0] / 784;                  // 16384
  int blocks = B / (WAVES_PER_BLOCK * 16);    // 64 images per block
  mnist_v3_kernel<<<blocks, WAVES_PER_BLOCK * 32, 0, stream>>>(
      image, W_ff, W_fb, W_vel, Wc1, bc1, Wc2, bc2, out);
}